// GATLinkPredictor_79920751444078
// MI455X (gfx1250) — compile-verified
//
#include <hip/hip_runtime.h>
#include <hip/hip_bf16.h>
#include <math.h>

typedef float v2f __attribute__((ext_vector_type(2)));
typedef float v8f __attribute__((ext_vector_type(8)));

#define NEG_SLOPE 0.2f

static __device__ __forceinline__ void atomicMaxF(float* addr, float v) {
    // Exact IEEE float max via integer atomics (valid with -inf init, no NaNs):
    // non-negative floats order like signed ints; negative floats order
    // reversed as unsigned ints.
    if (v >= 0.0f) {
        atomicMax((int*)addr, __float_as_int(v));
    } else {
        atomicMin((unsigned int*)addr, __float_as_uint(v));
    }
}

// ---------------------------------------------------------------------------
// Dense GEMM: H[nrows x 64] = X[nrows x K] * W[K x 64], fp32 via
// V_WMMA_F32_16X16X4_F32. 256 threads = 8 wave32; block covers 32 rows x 64
// cols = 8 (16x16) tiles, one per wave. X tile and whole W staged in LDS.
// ---------------------------------------------------------------------------
template <int K>
__global__ void __launch_bounds__(256)
gemm64_wmma_kernel(const float* __restrict__ X, const float* __restrict__ W,
                   float* __restrict__ H, int nrows) {
    constexpr int XS = K + 4;  // LDS row stride for X tile (stride-4 banks)
    constexpr int WS = 72;     // LDS row stride for W (disjoint half-wave banks)
    __shared__ float xs[32 * XS];
    __shared__ float ws[K * WS];

    const int tid = threadIdx.x;
    const int lane = tid & 31;
    const int wave = tid >> 5;
    const int rowBase = blockIdx.x * 32;

    // Stage W[K x 64] (coalesced)
    for (int idx = tid; idx < K * 64; idx += 256)
        ws[(idx >> 6) * WS + (idx & 63)] = W[idx];
    // Stage X tile [32 x K] (coalesced along K)
    for (int idx = tid; idx < 32 * K; idx += 256) {
        int r = idx / K, k = idx - r * K;
        int gr = rowBase + r;
        xs[r * XS + k] = (gr < nrows) ? X[(size_t)gr * K + k] : 0.0f;
    }
    __syncthreads();

    const int rT = wave >> 2;              // row tile 0..1
    const int cT = wave & 3;               // col tile 0..3
    const int mlo = lane & 15;             // M (A) / N (B,C,D) within tile
    const int khi = (lane >> 4) << 1;      // K offset: 0 (lanes 0-15), 2 (16-31)

    v8f c = {0.f, 0.f, 0.f, 0.f, 0.f, 0.f, 0.f, 0.f};
    const float* xrow = &xs[(rT * 16 + mlo) * XS];
    const float* wcol = &ws[cT * 16 + mlo];
#pragma unroll
    for (int kb = 0; kb < K; kb += 4) {
        v2f a, b;
        // A 16x4 layout: VGPR v -> K = kb + khi + v, M = lane%16
        a.x = xrow[kb + khi + 0];
        a.y = xrow[kb + khi + 1];
        // B 4x16 layout: VGPR v -> K = kb + khi + v, N = lane%16
        b.x = wcol[(kb + khi + 0) * WS];
        b.y = wcol[(kb + khi + 1) * WS];
        c = __builtin_amdgcn_wmma_f32_16x16x4_f32(
            /*neg_a=*/false, a, /*neg_b=*/false, b,
            /*c_mod=*/(short)0, c, /*reuse_a=*/false, /*reuse_b=*/false);
    }

    // C/D 16x16 f32: VGPR r -> M = r + 8*(lane>=16), N = lane%16
    const int outRowBase = rowBase + rT * 16 + ((lane >> 4) << 3);
    const int col = cT * 16 + mlo;
#pragma unroll
    for (int r = 0; r < 8; ++r) {
        int gr = outRowBase + r;
        if (gr < nrows) H[(size_t)gr * 64 + col] = c[r];
    }
}

// ---------------------------------------------------------------------------
// Init: out accumulators = 0, segment-max = -inf, denominators = 0
// ---------------------------------------------------------------------------
__global__ void __launch_bounds__(256)
init_kernel(float* m1, float* den1, float* out1,
            float* m2, float* den2, float* z, int N) {
    int t = blockIdx.x * blockDim.x + threadIdx.x;
    if (t < N * 64) { out1[t] = 0.0f; z[t] = 0.0f; }
    if (t < N * 8)  { m1[t] = -INFINITY; den1[t] = 0.0f; }
    if (t < N)      { m2[t] = -INFINITY; den2[t] = 0.0f; }
}

// ---------------------------------------------------------------------------
// Layer-1 attention scalars: alpha_s/alpha_d [N,8] from h1 [N,8,8]
// ---------------------------------------------------------------------------
__global__ void __launch_bounds__(256)
alpha1_kernel(const float* __restrict__ h1, const float* __restrict__ as1,
              const float* __restrict__ ad1, float* __restrict__ alpha_s,
              float* __restrict__ alpha_d, int N) {
    int t = blockIdx.x * blockDim.x + threadIdx.x;  // t = n*8 + head
    if (t >= N * 8) return;
    int h = t & 7;
    const float* hp = &h1[(size_t)t * 8];
    float s = 0.f, d = 0.f;
#pragma unroll
    for (int c = 0; c < 8; ++c) {
        float v = hp[c];
        s += v * as1[h * 8 + c];
        d += v * ad1[h * 8 + c];
    }
    alpha_s[t] = s;
    alpha_d[t] = d;
}

static __device__ __forceinline__ void edge_sd(const long long* __restrict__ esrc,
                                               const long long* __restrict__ edst,
                                               int e, int E, int& s, int& d) {
    if (e < E) { s = (int)esrc[e]; d = (int)edst[e]; }
    else       { s = d = e - E; }  // appended self-loops
}

// ---------------------------------------------------------------------------
// Layer 1 edge softmax (8 heads): max, then exp+denom, then weighted scatter
// ---------------------------------------------------------------------------
__global__ void __launch_bounds__(256)
edge_max1_kernel(const long long* __restrict__ esrc, const long long* __restrict__ edst,
                 int E, int ET, const float* __restrict__ as,
                 const float* __restrict__ ad, float* __restrict__ m) {
    int t = blockIdx.x * blockDim.x + threadIdx.x;  // t = e*8 + head
    if (t >= ET * 8) return;
    int e = t >> 3, h = t & 7, s, d;
    edge_sd(esrc, edst, e, E, s, d);
    float v = as[s * 8 + h] + ad[d * 8 + h];
    v = (v > 0.f) ? v : NEG_SLOPE * v;
    atomicMaxF(&m[d * 8 + h], v);
}

__global__ void __launch_bounds__(256)
edge_denom1_kernel(const long long* __restrict__ esrc, const long long* __restrict__ edst,
                   int E, int ET, const float* __restrict__ as,
                   const float* __restrict__ ad, const float* __restrict__ m,
                   float* __restrict__ den) {
    int t = blockIdx.x * blockDim.x + threadIdx.x;
    if (t >= ET * 8) return;
    int e = t >> 3, h = t & 7, s, d;
    edge_sd(esrc, edst, e, E, s, d);
    float v = as[s * 8 + h] + ad[d * 8 + h];
    v = (v > 0.f) ? v : NEG_SLOPE * v;
    atomicAdd(&den[d * 8 + h], expf(v - m[d * 8 + h]));
}

__global__ void __launch_bounds__(256)
edge_agg1_kernel(const long long* __restrict__ esrc, const long long* __restrict__ edst,
                 int E, int ET, const float* __restrict__ as,
                 const float* __restrict__ ad, const float* __restrict__ m,
                 const float* __restrict__ den, const float* __restrict__ h1,
                 float* __restrict__ out) {
    int t = blockIdx.x * blockDim.x + threadIdx.x;  // t = e*8 + head
    if (t >= ET * 8) return;
    int e = t >> 3, h = t & 7, s, d;
    edge_sd(esrc, edst, e, E, s, d);
    float v = as[s * 8 + h] + ad[d * 8 + h];
    v = (v > 0.f) ? v : NEG_SLOPE * v;
    float w = expf(v - m[d * 8 + h]) / (den[d * 8 + h] + 1e-16f);
    const float* hp = &h1[(size_t)s * 64 + h * 8];
    float* op = &out[(size_t)d * 64 + h * 8];
#pragma unroll
    for (int c = 0; c < 8; ++c) atomicAdd(&op[c], hp[c] * w);
}

// out1 <- elu(out1 + b1), in place
__global__ void __launch_bounds__(256)
bias_elu_kernel(float* __restrict__ out1, const float* __restrict__ b1, int N) {
    int t = blockIdx.x * blockDim.x + threadIdx.x;
    if (t >= N * 64) return;
    float v = out1[t] + b1[t & 63];
    out1[t] = (v > 0.f) ? v : (expf(v) - 1.f);
}

// ---------------------------------------------------------------------------
// Layer-2 attention scalars (1 head, 64 channels)
// ---------------------------------------------------------------------------
__global__ void __launch_bounds__(256)
alpha2_kernel(const float* __restrict__ h2, const float* __restrict__ as2,
              const float* __restrict__ ad2, float* __restrict__ asv,
              float* __restrict__ adv, int N) {
    int n = blockIdx.x * blockDim.x + threadIdx.x;
    if (n >= N) return;
    const float* hp = &h2[(size_t)n * 64];
    float s = 0.f, d = 0.f;
#pragma unroll
    for (int c = 0; c < 64; ++c) {
        float v = hp[c];
        s += v * as2[c];
        d += v * ad2[c];
    }
    asv[n] = s;
    adv[n] = d;
}

__global__ void __launch_bounds__(256)
edge_max2_kernel(const long long* __restrict__ esrc, const long long* __restrict__ edst,
                 int E, int ET, const float* __restrict__ as,
                 const float* __restrict__ ad, float* __restrict__ m) {
    int e = blockIdx.x * blockDim.x + threadIdx.x;
    if (e >= ET) return;
    int s, d;
    edge_sd(esrc, edst, e, E, s, d);
    float v = as[s] + ad[d];
    v = (v > 0.f) ? v : NEG_SLOPE * v;
    atomicMaxF(&m[d], v);
}

__global__ void __launch_bounds__(256)
edge_denom2_kernel(const long long* __restrict__ esrc, const long long* __restrict__ edst,
                   int E, int ET, const float* __restrict__ as,
                   const float* __restrict__ ad, const float* __restrict__ m,
                   float* __restrict__ den) {
    int e = blockIdx.x * blockDim.x + threadIdx.x;
    if (e >= ET) return;
    int s, d;
    edge_sd(esrc, edst, e, E, s, d);
    float v = as[s] + ad[d];
    v = (v > 0.f) ? v : NEG_SLOPE * v;
    atomicAdd(&den[d], expf(v - m[d]));
}

__global__ void __launch_bounds__(256)
edge_agg2_kernel(const long long* __restrict__ esrc, const long long* __restrict__ edst,
                 int E, int ET, const float* __restrict__ as,
                 const float* __restrict__ ad, const float* __restrict__ m,
                 const float* __restrict__ den, const float* __restrict__ h2,
                 float* __restrict__ z) {
    int t = blockIdx.x * blockDim.x + threadIdx.x;  // t = e*4 + quarter
    if (t >= ET * 4) return;
    int e = t >> 2, c0 = (t & 3) << 4, s, d;
    edge_sd(esrc, edst, e, E, s, d);
    float v = as[s] + ad[d];
    v = (v > 0.f) ? v : NEG_SLOPE * v;
    float w = expf(v - m[d]) / (den[d] + 1e-16f);
    const float* hp = &h2[(size_t)s * 64 + c0];
    float* zp = &z[(size_t)d * 64 + c0];
#pragma unroll
    for (int c = 0; c < 16; ++c) atomicAdd(&zp[c], hp[c] * w);
}

__global__ void __launch_bounds__(256)
zbias_kernel(float* __restrict__ z, const float* __restrict__ b2, int N) {
    int t = blockIdx.x * blockDim.x + threadIdx.x;
    if (t >= N * 64) return;
    z[t] += b2[t & 63];
}

// decode: sigmoid(dot(z[q0], z[q1]))
__global__ void __launch_bounds__(256)
decode_kernel(const long long* __restrict__ qe, int Q,
              const float* __restrict__ z, float* __restrict__ out) {
    int q = blockIdx.x * blockDim.x + threadIdx.x;
    if (q >= Q) return;
    const float* za = &z[(size_t)qe[q] * 64];
    const float* zb = &z[(size_t)qe[Q + q] * 64];
    float s = 0.f;
#pragma unroll
    for (int c = 0; c < 64; ++c) s += za[c] * zb[c];
    out[q] = 1.0f / (1.0f + expf(-s));
}

static inline int cdiv(long long a, int b) { return (int)((a + b - 1) / b); }

extern "C" void kernel_launch(void* const* d_in, const int* in_sizes, int n_in,
                              void* d_out, int out_size, void* d_ws, size_t ws_size,
                              hipStream_t stream) {
    const float*     x    = (const float*)d_in[0];
    const long long* ei   = (const long long*)d_in[1];  // int64 [2,E]
    const long long* qe   = (const long long*)d_in[2];  // int64 [2,Q]
    const float*     W1   = (const float*)d_in[3];
    const float*     as1  = (const float*)d_in[4];
    const float*     ad1  = (const float*)d_in[5];
    const float*     b1   = (const float*)d_in[6];
    const float*     W2   = (const float*)d_in[7];
    const float*     as2  = (const float*)d_in[8];
    const float*     ad2  = (const float*)d_in[9];
    const float*     b2   = (const float*)d_in[10];
    float* out = (float*)d_out;

    const int N = in_sizes[0] / 128;
    const int E = in_sizes[1] / 2;
    const int Q = in_sizes[2] / 2;
    const int ET = E + N;  // edges + self-loops
    const long long* esrc = ei;
    const long long* edst = ei + E;

    // Workspace layout (floats): ~N*292 (N*1168 bytes)
    float* ws   = (float*)d_ws;
    float* h1   = ws;                       // N*64
    float* as1v = h1   + (size_t)N * 64;    // N*8
    float* ad1v = as1v + (size_t)N * 8;     // N*8
    float* m1   = ad1v + (size_t)N * 8;     // N*8
    float* den1 = m1   + (size_t)N * 8;     // N*8
    float* out1 = den1 + (size_t)N * 8;     // N*64 (becomes ELU'd hidden)
    float* h2   = out1 + (size_t)N * 64;    // N*64
    float* as2v = h2   + (size_t)N * 64;    // N
    float* ad2v = as2v + (size_t)N;         // N
    float* m2   = ad2v + (size_t)N;         // N
    float* den2 = m2   + (size_t)N;         // N
    float* z    = den2 + (size_t)N;         // N*64

    const int B = 256;

    // 0) init accumulators
    init_kernel<<<cdiv((long long)N * 64, B), B, 0, stream>>>(m1, den1, out1, m2, den2, z, N);

    // 1) h1 = x @ W1  (WMMA fp32, K=128)
    gemm64_wmma_kernel<128><<<cdiv(N, 32), B, 0, stream>>>(x, W1, h1, N);

    // 2) per-node attention scalars
    alpha1_kernel<<<cdiv((long long)N * 8, B), B, 0, stream>>>(h1, as1, ad1, as1v, ad1v, N);

    // 3) segment softmax + aggregation (layer 1, 8 heads)
    edge_max1_kernel<<<cdiv((long long)ET * 8, B), B, 0, stream>>>(esrc, edst, E, ET, as1v, ad1v, m1);
    edge_denom1_kernel<<<cdiv((long long)ET * 8, B), B, 0, stream>>>(esrc, edst, E, ET, as1v, ad1v, m1, den1);
    edge_agg1_kernel<<<cdiv((long long)ET * 8, B), B, 0, stream>>>(esrc, edst, E, ET, as1v, ad1v, m1, den1, h1, out1);

    // 4) +bias, ELU (in place)
    bias_elu_kernel<<<cdiv((long long)N * 64, B), B, 0, stream>>>(out1, b1, N);

    // 5) h2 = hidden @ W2  (WMMA fp32, K=64)
    gemm64_wmma_kernel<64><<<cdiv(N, 32), B, 0, stream>>>(out1, W2, h2, N);

    // 6) layer-2 attention scalars (1 head)
    alpha2_kernel<<<cdiv(N, B), B, 0, stream>>>(h2, as2, ad2, as2v, ad2v, N);

    // 7) segment softmax + aggregation (layer 2)
    edge_max2_kernel<<<cdiv(ET, B), B, 0, stream>>>(esrc, edst, E, ET, as2v, ad2v, m2);
    edge_denom2_kernel<<<cdiv(ET, B), B, 0, stream>>>(esrc, edst, E, ET, as2v, ad2v, m2, den2);
    edge_agg2_kernel<<<cdiv((long long)ET * 4, B), B, 0, stream>>>(esrc, edst, E, ET, as2v, ad2v, m2, den2, h2, z);

    // 8) + b2, then dot-product decode with sigmoid
    zbias_kernel<<<cdiv((long long)N * 64, B), B, 0, stream>>>(z, b2, N);
    decode_kernel<<<cdiv(Q, B), B, 0, stream>>>(qe, Q, z, out);
}